// HaarDown_11682311045469
// MI455X (gfx1250) — compile-verified
//
#include <hip/hip_runtime.h>

typedef float v2f __attribute__((ext_vector_type(2)));
typedef float v8f __attribute__((ext_vector_type(8)));

// x: (8, 64, 512, 512) f32  ->  out: (8, 192, 256, 256) f32
// out[b, 3c+k, h2, w2] = M[k] . [a,b,c,d], patch at (2h2, 2w2):
//   M[0] = 0.5*[ 1,-1,-1, 1]
//   M[1] = 0.5*[ 1,-1, 1,-1]
//   M[2] = 0.5*[ 1, 1,-1,-1]
//
// Per 32-pixel tile: two chained V_WMMA_F32_16X16X4_F32.
//   A1 carries M in rows 0..2  -> results land in lanes 0..15 of D VGPRs 0..2
//   A2 carries M in rows 8..10 -> results land in lanes 16..31 of D VGPRs 0..2
//   d = WMMA(A1, B_pix0_15, WMMA(A2, B_pix16_31, 0))
// => D VGPR k is o_k for all 32 pixels, already in coalesced store order.
// B operands come from one coalesced float2 load each: lanes 0-15 read {a,b}
// from row 2h2, lanes 16-31 read {c,d} from row 2h2+1.
// Each wave processes 4 consecutive row-pairs (tiles) with immediate offsets.

__global__ __launch_bounds__(256) void haar_down_wmma(const float* __restrict__ x,
                                                      float* __restrict__ out) {
  const unsigned lane = threadIdx.x & 31u;
  const unsigned wave = blockIdx.x * 8u + (threadIdx.x >> 5);

  const unsigned wgroup = wave & 7u;    // 32-pixel group along w2 (8 per row)
  const unsigned tmp    = wave >> 3;
  const unsigned h2b    = tmp & 63u;    // block of 4 output rows
  const unsigned bc     = tmp >> 6;     // b*64 + c, in [0, 512)

  // ---- A matrices (one-time, loop-invariant) ----
  // A VGPR layout (16x4 f32): lanes 0-15 -> (K0,K1) of row M=lane,
  //                           lanes 16-31 -> (K2,K3) of row M=lane-16.
  const int  m  = (int)(lane & 15u);
  const bool hi = lane >= 16u;

  // Haar row s (s=0,1,2): (K0,K1) / (K2,K3) values.
  //   s=0: lo ( .5,-.5)  hi (-.5, .5)
  //   s=1: lo ( .5,-.5)  hi ( .5,-.5)
  //   s=2: lo ( .5, .5)  hi (-.5,-.5)
#define MK_A(sv, AX, AY)                                              \
  {                                                                   \
    const int s = (sv);                                               \
    AX = 0.0f; AY = 0.0f;                                             \
    AX = (s == 0) ? (hi ? -0.5f :  0.5f) : AX;                        \
    AY = (s == 0) ? (hi ?  0.5f : -0.5f) : AY;                        \
    AX = (s == 1) ?  0.5f                : AX;                        \
    AY = (s == 1) ? -0.5f                : AY;                        \
    AX = (s == 2) ? (hi ? -0.5f :  0.5f) : AX;                        \
    AY = (s == 2) ? (hi ? -0.5f :  0.5f) : AY;                        \
  }

  float a1x, a1y, a2x, a2y;
  MK_A(m,     a1x, a1y)   // rows 0..2
  MK_A(m - 8, a2x, a2y)   // rows 8..10
  v2f amat1 = {a1x, a1y};
  v2f amat2 = {a2x, a2y};

  // ---- base addresses (tile 0) ----
  const unsigned half = lane >> 4;  // 0: row 2h2 ({a,b}), 1: row 2h2+1 ({c,d})
  const float* pin = x
      + ((size_t)bc * 512u + 8u * h2b + half) * 512u
      + wgroup * 64u + (unsigned)m * 2u;
  float* pout = out
      + ((size_t)(bc * 3u) * 256u + 4u * h2b) * 256u
      + wgroup * 32u + lane;

  // ---- 4 tiles: per tile 2 NT loads, 2 chained WMMAs, 3 NT stores ----
  const v8f zero = {};
#pragma unroll
  for (int t = 0; t < 4; ++t) {
    const v2f b_lo = __builtin_nontemporal_load(
        (const v2f*)(pin + t * 1024));          // pixels 0-15  (t*2 rows ahead)
    const v2f b_hi = __builtin_nontemporal_load(
        (const v2f*)(pin + t * 1024 + 32));     // pixels 16-31

    v8f acc = __builtin_amdgcn_wmma_f32_16x16x4_f32(
        false, amat2, false, b_hi, (short)0, zero, false, false);
    acc = __builtin_amdgcn_wmma_f32_16x16x4_f32(
        false, amat1, false, b_lo, (short)0, acc, false, false);

#pragma unroll
    for (int k = 0; k < 3; ++k)
      __builtin_nontemporal_store(acc[k], pout + t * 256 + k * 65536);
  }
#undef MK_A
}

extern "C" void kernel_launch(void* const* d_in, const int* in_sizes, int n_in,
                              void* d_out, int out_size, void* d_ws, size_t ws_size,
                              hipStream_t stream) {
  (void)in_sizes; (void)n_in; (void)out_size; (void)d_ws; (void)ws_size;
  const float* x = (const float*)d_in[0];
  float* out = (float*)d_out;

  // 512 (b*c) * 64 (h2 blocks of 4) * 8 (w groups) = 262,144 waves; 8/block.
  const unsigned blocks = 512u * 64u * 8u / 8u;  // 32768
  haar_down_wmma<<<dim3(blocks), dim3(256), 0, stream>>>(x, out);
}